// S5Kernel_68521908240464
// MI455X (gfx1250) — compile-verified
//
#include <hip/hip_runtime.h>
#include <stdint.h>

// S5 associative scan, T=131072, D=256, fp32.
// Memory-bound (536MB min traffic @ 23.3TB/s ~ 23us). Strategy:
//  K1: per-chunk aggregates (async global->LDS streaming, wave32, 256 ch/block)
//  K2: exclusive scan of chunk aggregates (1 block)
//  K3: re-stream + local scan + apply prefix, coalesced stores.

#define T_STEPS 131072
#define DDIM    256
#define NCHUNK  1024
#define CHUNK   128                 // T_STEPS / NCHUNK
#define TILE_R  16                  // rows per LDS tile
#define NTILE   (CHUNK / TILE_R)    // 8 tiles per chunk
#define TILE_F  (TILE_R * DDIM)     // 4096 floats = 16KB per array per buffer

typedef int v4i __attribute__((vector_size(16)));
typedef __attribute__((address_space(1))) v4i gv4i;   // global (prints as __device__)
typedef __attribute__((address_space(3))) v4i lv4i;   // LDS
typedef __attribute__((address_space(3))) void lds_void;

// ---- CDNA5 async global->LDS copy (16B per lane, wave-wide) ----------------
__device__ __forceinline__ void async_load16(const float* gsrc, float* ldsdst) {
#if __has_builtin(__builtin_amdgcn_global_load_async_to_lds_b128)
    __builtin_amdgcn_global_load_async_to_lds_b128(
        (gv4i*)gsrc, (lv4i*)ldsdst, /*offset=*/0, /*cpol=*/0);
#else
    uint32_t loff = (uint32_t)(uintptr_t)(lds_void*)ldsdst;
    uint64_t gaddr = (uint64_t)(uintptr_t)gsrc;
    asm volatile("global_load_async_to_lds_b128 %0, %1, off"
                 :: "v"(loff), "v"(gaddr) : "memory");
#endif
}

template <int N>
__device__ __forceinline__ void wait_async() {
#if __has_builtin(__builtin_amdgcn_s_wait_asynccnt)
    __builtin_amdgcn_s_wait_asynccnt(N);
#else
    asm volatile("s_wait_asynccnt %0" :: "i"(N));
#endif
}

// Issue one 16-row x 256-ch tile of A and Bu into LDS buffer `buf`.
// Per wave: 4 sweeps x 2 arrays = 8 async instructions (ASYNCcnt <= 16 dbl-buf).
__device__ __forceinline__ void issue_tile(const float* __restrict__ gA,
                                           const float* __restrict__ gBu,
                                           float* sA, float* sB,
                                           int row0, int t, int buf, int tid) {
    const int gbase = (row0 + t * TILE_R) * DDIM;   // tile is contiguous in memory
#pragma unroll
    for (int s = 0; s < 4; ++s) {
        const int off = tid * 4 + s * 1024;          // float offset within tile
        async_load16(gA  + gbase + off, sA + buf * TILE_F + off);
        async_load16(gBu + gbase + off, sB + buf * TILE_F + off);
    }
}

// Streamed serial scan of one chunk. Thread `tid` owns channel `tid`.
// If STORE: writes outA/outB with prefix (pA,pB) applied.
// Returns chunk aggregate in (rA,rB).
template <bool STORE>
__device__ __forceinline__ void chunk_scan(const float* __restrict__ gA,
                                           const float* __restrict__ gBu,
                                           float* __restrict__ outA,
                                           float* __restrict__ outB,
                                           float pA, float pB,
                                           float& rA, float& rB,
                                           float* sA, float* sB,
                                           int c, int tid) {
    const int row0 = c * CHUNK;
    issue_tile(gA, gBu, sA, sB, row0, 0, 0, tid);

    float Al = 1.0f, Bl = 0.0f;
    for (int t = 0; t < NTILE; ++t) {
        const int buf = t & 1;
        if (t + 1 < NTILE) {
            issue_tile(gA, gBu, sA, sB, row0, t + 1, buf ^ 1, tid);
            wait_async<8>();          // previous 8 (current tile) complete in-order
        } else {
            wait_async<0>();
        }
        __syncthreads();              // tile written by all waves -> visible

        const float* tA = sA + buf * TILE_F;
        const float* tB = sB + buf * TILE_F;
#pragma unroll
        for (int r = 0; r < TILE_R; ++r) {
            const float a = tA[r * DDIM + tid];   // conflict-free: row*256+tid
            const float b = tB[r * DDIM + tid];
            Bl = a * Bl + b;
            Al = a * Al;
            if (STORE) {
                const int row = row0 + t * TILE_R + r;
                outA[(size_t)row * DDIM + tid] = Al * pA;
                outB[(size_t)row * DDIM + tid] = Al * pB + Bl;
            }
        }
        __syncthreads();              // everyone done reading before buf reuse
    }
    rA = Al;
    rB = Bl;
}

// ---- Kernel 1: per-chunk aggregates ----------------------------------------
__global__ __launch_bounds__(256) void s5_chunk_agg(const float* __restrict__ A,
                                                    const float* __restrict__ Bu,
                                                    float* __restrict__ wsA,
                                                    float* __restrict__ wsB) {
    __shared__ float sA[2 * TILE_F];
    __shared__ float sB[2 * TILE_F];
    const int c = blockIdx.x, tid = threadIdx.x;
    float aA, aB;
    chunk_scan<false>(A, Bu, nullptr, nullptr, 1.0f, 0.0f, aA, aB, sA, sB, c, tid);
    wsA[c * DDIM + tid] = aA;
    wsB[c * DDIM + tid] = aB;
}

// ---- Kernel 2: exclusive scan of chunk aggregates (1 block) ----------------
__global__ __launch_bounds__(256) void s5_prefix(const float* __restrict__ wsA,
                                                 const float* __restrict__ wsB,
                                                 float* __restrict__ pfxA,
                                                 float* __restrict__ pfxB) {
    const int d = threadIdx.x;
    float pA = 1.0f, pB = 0.0f;
    for (int c = 0; c < NCHUNK; ++c) {
        pfxA[c * DDIM + d] = pA;
        pfxB[c * DDIM + d] = pB;
        const float a = wsA[c * DDIM + d];
        const float b = wsB[c * DDIM + d];
        pB = a * pB + b;
        pA = a * pA;
    }
}

// ---- Kernel 3: recompute local scan, apply prefix, store -------------------
__global__ __launch_bounds__(256) void s5_apply(const float* __restrict__ A,
                                                const float* __restrict__ Bu,
                                                const float* __restrict__ pfxA,
                                                const float* __restrict__ pfxB,
                                                float* __restrict__ outA,
                                                float* __restrict__ outB) {
    __shared__ float sA[2 * TILE_F];
    __shared__ float sB[2 * TILE_F];
    const int c = blockIdx.x, tid = threadIdx.x;
    const float pA = pfxA[c * DDIM + tid];
    const float pB = pfxB[c * DDIM + tid];
    float aA, aB;
    chunk_scan<true>(A, Bu, outA, outB, pA, pB, aA, aB, sA, sB, c, tid);
}

extern "C" void kernel_launch(void* const* d_in, const int* in_sizes, int n_in,
                              void* d_out, int out_size, void* d_ws, size_t ws_size,
                              hipStream_t stream) {
    (void)in_sizes; (void)n_in; (void)out_size; (void)ws_size;
    const float* A  = (const float*)d_in[0];
    const float* Bu = (const float*)d_in[1];
    float* outA = (float*)d_out;                       // tuple order: (A, Bu)
    float* outB = outA + (size_t)T_STEPS * DDIM;

    float* ws   = (float*)d_ws;                        // 4 * 1024 * 256 * 4B = 4MB
    float* wsA  = ws;
    float* wsB  = ws + 1 * NCHUNK * DDIM;
    float* pfxA = ws + 2 * NCHUNK * DDIM;
    float* pfxB = ws + 3 * NCHUNK * DDIM;

    s5_chunk_agg<<<NCHUNK, 256, 0, stream>>>(A, Bu, wsA, wsB);
    s5_prefix<<<1, 256, 0, stream>>>(wsA, wsB, pfxA, pfxB);
    s5_apply<<<NCHUNK, 256, 0, stream>>>(A, Bu, pfxA, pfxB, outA, outB);
}